// QueryAndGroupQuat_35725537968264
// MI455X (gfx1250) — compile-verified
//
#include <hip/hip_runtime.h>
#include <hip/hip_bf16.h>
#include <math.h>

// ---------------------------------------------------------------------------
// QueryAndGroupQuat for MI455X (gfx1250, wave32)
//
//   B=4, N=8192, S=2048 queries, K=32 samples, RADIUS=0.1, M=8 rolls
//
// Kernel 1: WMMA-based ball query.
//   One wave handles 16 queries; streams the 8192 points 32 at a time with
//   TWO back-to-back V_WMMA_F32_16X16X4_F32 per iteration:
//     A (16x4) row m  = (qx, qy, qz, 1)
//     B (4x16) col n  = (-2*xx, -2*xy, -2*xz, |x|^2)
//     C           row = |q_m|^2 broadcast
//   => D = |q|^2 + |x|^2 - 2 q.x  (exact f32 squared distances, 256/WMMA).
//   D tile layout (ISA 7.12.2): VGPR r -> rows {r, r+8}, lanes -> cols, so
//   __ballot(d[r] < R2) packs row r's 16-bit mask in bits[15:0] and row
//   r+8's in bits[31:16]. A branchless cndmask chain routes the right mask
//   to the lane that owns each query row; lanes 0..15 then peel set bits
//   (ctz) collecting the first 33 in-radius indices in ascending order
//   (pointnet2 semantics), filling unused slots with the first hit.
//
// Kernel 2: group + rot-sort + quaternion + rolls.
//   One wave per query, lane = neighbor. Angle in the plane orthogonal to
//   the center direction, 32-lane bitonic sort via __shfl_xor with
//   (angle, lane) keys (stable-argsort emulation), quaternion, then 8 rolls
//   via __shfl(q,(lane+m)&31). All 32-float output rows are lane-coalesced.
// ---------------------------------------------------------------------------

#define BB       4
#define NN       8192
#define SS       2048
#define KK       32          // NSAMPLE
#define KP1      33          // NSAMPLE + 1
#define MM       8
#define RADIUS   0.1f
#define R2       (RADIUS * RADIUS)
#define PI_F     3.14159265358979323846f

typedef float v2f __attribute__((ext_vector_type(2)));
typedef float v8f __attribute__((ext_vector_type(8)));

// branchless 8-way select: executed by ALL lanes -> lowers to v_cndmask chain
__device__ __forceinline__ unsigned sel8(int r,
                                         unsigned b0, unsigned b1,
                                         unsigned b2, unsigned b3,
                                         unsigned b4, unsigned b5,
                                         unsigned b6, unsigned b7)
{
    unsigned v = b0;
    v = (r == 1) ? b1 : v;
    v = (r == 2) ? b2 : v;
    v = (r == 3) ? b3 : v;
    v = (r == 4) ? b4 : v;
    v = (r == 5) ? b5 : v;
    v = (r == 6) ? b6 : v;
    v = (r == 7) ? b7 : v;
    return v;
}

// ---------------------------------------------------------------------------
// Kernel 1: ball query. grid = B*(S/16) blocks of 32 threads (1 wave).
// idxbuf layout: int idxbuf[B][S][33]
// ---------------------------------------------------------------------------
__global__ __launch_bounds__(32)
void ball_query_wmma_kernel(const float* __restrict__ xyz,
                            const float* __restrict__ new_xyz,
                            int* __restrict__ idxbuf)
{
    const int blocksPerBatch = SS / 16;
    const int b     = blockIdx.x / blocksPerBatch;
    const int qbase = (blockIdx.x % blocksPerBatch) * 16;

    const int lane = threadIdx.x;        // 0..31
    const int half = lane >> 4;          // 0: lanes 0-15, 1: lanes 16-31
    const int m16  = lane & 15;          // row / point-in-chunk id

    // ---- query fragment (constant over the whole N loop) ----
    const float* q = &new_xyz[((size_t)b * SS + qbase + m16) * 3];
    const float qx = q[0], qy = q[1], qz = q[2];
    const float qq = qx * qx + qy * qy + qz * qz;

    // A 16x4 layout: VGPR0 = K0 (lanes 0-15) / K2 (lanes 16-31), VGPR1 = K1/K3
    v2f afrag;
    afrag[0] = half ? qz : qx;           // K2 : K0
    afrag[1] = half ? 1.0f : qy;         // K3 : K1

    // C tile: VGPR r holds |q_r|^2 (lanes 0-15) and |q_{r+8}|^2 (lanes 16-31)
    v8f cfrag;
#pragma unroll
    for (int r = 0; r < 8; ++r)
        cfrag[r] = __shfl(qq, r + (half << 3), 32);

    int cnt   = (lane < 16) ? 0 : KP1;   // idle lanes pretend "done"
    int first = 0;
    const int    r8   = lane & 7;
    const size_t qoff = ((size_t)b * SS + qbase + m16) * KP1;
    const float* xb   = &xyz[(size_t)b * NN * 3];

    for (int base = 0; base < NN; base += 32) {
        // prefetch the next 32-point block (global_prefetch_b8)
        if (base + 32 < NN)
            __builtin_prefetch(&xb[(base + 32 + lane) * 3], 0, 0);

        // ---- point fragments: both halves read the same 16 points ----
        const float* p0 = &xb[(base + m16) * 3];
        const float p0x = p0[0], p0y = p0[1], p0z = p0[2];
        v2f bf0;
        bf0[0] = half ? (-2.0f * p0z) : (-2.0f * p0x);
        bf0[1] = half ? (p0x * p0x + p0y * p0y + p0z * p0z) : (-2.0f * p0y);

        const float* p1 = &xb[(base + 16 + m16) * 3];
        const float p1x = p1[0], p1y = p1[1], p1z = p1[2];
        v2f bf1;
        bf1[0] = half ? (-2.0f * p1z) : (-2.0f * p1x);
        bf1[1] = half ? (p1x * p1x + p1y * p1y + p1z * p1z) : (-2.0f * p1y);

        // ---- two d^2 tiles: 512 distance tests per iteration ----
        v8f d0 = __builtin_amdgcn_wmma_f32_16x16x4_f32(
            false, afrag, false, bf0, (short)0, cfrag, false, false);
        v8f d1 = __builtin_amdgcn_wmma_f32_16x16x4_f32(
            false, afrag, false, bf1, (short)0, cfrag, false, false);

        // ---- tiles -> per-row in-radius masks (wave-uniform ballots) ----
        const unsigned A0 = (unsigned)__ballot(d0[0] < R2);
        const unsigned A1 = (unsigned)__ballot(d0[1] < R2);
        const unsigned A2 = (unsigned)__ballot(d0[2] < R2);
        const unsigned A3 = (unsigned)__ballot(d0[3] < R2);
        const unsigned A4 = (unsigned)__ballot(d0[4] < R2);
        const unsigned A5 = (unsigned)__ballot(d0[5] < R2);
        const unsigned A6 = (unsigned)__ballot(d0[6] < R2);
        const unsigned A7 = (unsigned)__ballot(d0[7] < R2);
        const unsigned C0 = (unsigned)__ballot(d1[0] < R2);
        const unsigned C1 = (unsigned)__ballot(d1[1] < R2);
        const unsigned C2 = (unsigned)__ballot(d1[2] < R2);
        const unsigned C3 = (unsigned)__ballot(d1[3] < R2);
        const unsigned C4 = (unsigned)__ballot(d1[4] < R2);
        const unsigned C5 = (unsigned)__ballot(d1[5] < R2);
        const unsigned C6 = (unsigned)__ballot(d1[6] < R2);
        const unsigned C7 = (unsigned)__ballot(d1[7] < R2);

        // ---- branchless routing: lane m gets row m's 16-bit mask ----
        const unsigned bal0 = sel8(r8, A0, A1, A2, A3, A4, A5, A6, A7);
        const unsigned bal1 = sel8(r8, C0, C1, C2, C3, C4, C5, C6, C7);
        unsigned mask0 = (lane < 8) ? (bal0 & 0xFFFFu) : (bal0 >> 16);
        unsigned mask1 = (lane < 8) ? (bal1 & 0xFFFFu) : (bal1 >> 16);

        // ---- sequential first-33 collection (lanes >=16 have cnt==33) ----
        if (cnt < KP1) {
            while (mask0 && cnt < KP1) {
                const int pid = base + __builtin_ctz(mask0);
                if (cnt == 0) first = pid;
                idxbuf[qoff + cnt] = pid;
                ++cnt;
                mask0 &= mask0 - 1;
            }
            while (mask1 && cnt < KP1) {
                const int pid = base + 16 + __builtin_ctz(mask1);
                if (cnt == 0) first = pid;
                idxbuf[qoff + cnt] = pid;
                ++cnt;
                mask1 &= mask1 - 1;
            }
        }
        if (__all(cnt >= KP1)) break;    // uniform early exit
    }

    // fill remaining slots with the first found index (0 if none found)
    if (lane < 16) {
        for (int t = cnt; t < KP1; ++t)
            idxbuf[qoff + t] = first;
    }
}

// ---------------------------------------------------------------------------
// deterministic hash -> uniform angle in (-pi, pi) for degenerate projections
// ---------------------------------------------------------------------------
__device__ __forceinline__ float hash_angle(unsigned v)
{
    v *= 2654435761u;
    v ^= v >> 16;
    v *= 2246822519u;
    v ^= v >> 13;
    const float u = (float)(v >> 8) * (1.0f / 16777216.0f);  // [0,1)
    return (u - 0.5f) * (2.0f * PI_F);
}

// ---------------------------------------------------------------------------
// Kernel 2: group + rot-sort + quat + rolls. One wave per query.
// grid = B*S/8 blocks of 256 threads (8 waves).
// out layout: float out[B][32][S][32], channel = c*8 + m (c in 0..3, m roll)
// ---------------------------------------------------------------------------
__global__ __launch_bounds__(256)
void group_sort_quat_kernel(const float* __restrict__ xyz,
                            const float* __restrict__ new_xyz,
                            const int* __restrict__ idxbuf,
                            float* __restrict__ out)
{
    const int wid  = blockIdx.x * 8 + (threadIdx.x >> 5);
    const int lane = threadIdx.x & 31;
    const int b    = wid / SS;
    const int s    = wid % SS;

    // ---- center & per-query reference frame (uniform across the wave) ----
    const float* c = &new_xyz[((size_t)b * SS + s) * 3];
    const float cx = c[0], cy = c[1], cz = c[2];
    const float cn   = sqrtf(cx * cx + cy * cy + cz * cz);
    const float cinv = 1.0f / (cn + 1e-6f);
    const float pnx = cx * cinv, pny = cy * cinv, pnz = cz * cinv;

    float rx, ry, rz;
    if (fabsf(pnx) > 0.999f) {           // colinear with x-axis -> project e_y
        rx = -pny * pnx; ry = 1.0f - pny * pny; rz = -pny * pnz;
    } else {                             // project e_x
        rx = 1.0f - pnx * pnx; ry = -pnx * pny; rz = -pnx * pnz;
    }
    const float rn   = sqrtf(rx * rx + ry * ry + rz * rz);
    const float rinv = 1.0f / (rn + 1e-6f);
    rx *= rinv; ry *= rinv; rz *= rinv;

    // ---- gather my neighbor (slots 1..32 of the 33 collected) ----
    const int nidx = idxbuf[((size_t)b * SS + s) * KP1 + 1 + lane];
    const float* pt = &xyz[((size_t)b * NN + nidx) * 3];
    float vx = pt[0] - cx, vy = pt[1] - cy, vz = pt[2] - cz;

    // ---- angle of projection onto plane orthogonal to p ----
    const float dotpv = pnx * vx + pny * vy + pnz * vz;
    float ux = vx - dotpv * pnx;
    float uy = vy - dotpv * pny;
    float uz = vz - dotpv * pnz;
    const float un   = sqrtf(ux * ux + uy * uy + uz * uz);
    const float uinv = 1.0f / (un + 1e-6f);
    ux *= uinv; uy *= uinv; uz *= uinv;
    const float pp2 = ux * ux + uy * uy + uz * uz;

    const float crx = ry * uz - rz * uy;
    const float cry = rz * ux - rx * uz;
    const float crz = rx * uy - ry * ux;
    const float sinus   = crx * pnx + cry * pny + crz * pnz;
    const float cosinus = rx * ux + ry * uy + rz * uz;
    float ang = atan2f(sinus, cosinus);
    if (pp2 < 1e-12f)
        ang = hash_angle((unsigned)(wid * 37 + lane));

    // ---- 32-lane bitonic sort on (ang, key); key tie-break => stable ----
    int key = lane;
#pragma unroll
    for (int size = 2; size <= 32; size <<= 1) {
#pragma unroll
        for (int stride = size >> 1; stride > 0; stride >>= 1) {
            const float oang = __shfl_xor(ang, stride, 32);
            const int   okey = __shfl_xor(key, stride, 32);
            const bool ascending = ((lane & size) == 0) || (size == 32);
            const bool lower     = ((lane & stride) == 0);
            const bool keepMin   = (ascending == lower);
            const bool selfGreater =
                (ang > oang) || (ang == oang && key > okey);
            const bool takeOther = keepMin ? selfGreater : !selfGreater;
            if (takeOther) { ang = oang; key = okey; }
        }
    }

    // pull sorted neighbor's vector from its original lane
    const float svx = __shfl(vx, key, 32);
    const float svy = __shfl(vy, key, 32);
    const float svz = __shfl(vz, key, 32);

    // ---- quaternion ----
    const float dist = sqrtf(svx * svx + svy * svy + svz * svz);
    const float oinv = 1.0f / (dist + 1e-6f);
    const float theta = dist * (PI_F * 0.5f / RADIUS);
    const float qw = cosf(theta);
    const float sf = sinf(theta) * oinv;
    const float q1 = sf * svx, q2 = sf * svy, q3 = sf * svz;

    // ---- M=8 rolls; out[b][c*8+m][s][k], roll -m => take (k+m)%32 ----
#pragma unroll
    for (int m = 0; m < MM; ++m) {
        const int srcl = (lane + m) & 31;
        const float w = __shfl(qw, srcl, 32);
        const float x = __shfl(q1, srcl, 32);
        const float y = __shfl(q2, srcl, 32);
        const float z = __shfl(q3, srcl, 32);
        const size_t rowBase = (size_t)b * (4 * MM);
        out[(((rowBase + 0 * MM + m) * SS) + s) * KK + lane] = w;
        out[(((rowBase + 1 * MM + m) * SS) + s) * KK + lane] = x;
        out[(((rowBase + 2 * MM + m) * SS) + s) * KK + lane] = y;
        out[(((rowBase + 3 * MM + m) * SS) + s) * KK + lane] = z;
    }
}

// ---------------------------------------------------------------------------
extern "C" void kernel_launch(void* const* d_in, const int* in_sizes, int n_in,
                              void* d_out, int out_size, void* d_ws, size_t ws_size,
                              hipStream_t stream)
{
    const float* xyz     = (const float*)d_in[0];   // (B, N, 3) f32
    const float* new_xyz = (const float*)d_in[1];   // (B, S, 3) f32
    float*       out     = (float*)d_out;           // (B, 4*M, S, K) f32
    int*         idxbuf  = (int*)d_ws;              // B*S*33 ints (~1.06 MB)

    (void)in_sizes; (void)n_in; (void)out_size; (void)ws_size;

    // Kernel 1: one wave per 16 queries
    ball_query_wmma_kernel<<<dim3(BB * (SS / 16)), dim3(32), 0, stream>>>(
        xyz, new_xyz, idxbuf);

    // Kernel 2: one wave per query, 8 waves per block
    group_sort_quat_kernel<<<dim3(BB * SS / 8), dim3(256), 0, stream>>>(
        xyz, new_xyz, idxbuf, out);
}